// CliffordSteerableKernel_50457275794056
// MI455X (gfx1250) — compile-verified
//
#include <hip/hip_runtime.h>
#include <math.h>

#define NB 8
#define NPOS 343
#define MP 352          // 343 padded to multiple of 32
#define PSUM 20
#define EPSF 1e-6f
#define SQ_EPSF 1e-12f
#define FACTORF (20.0f / 49.0f)   // 20 / KERNEL_SIZE^(DIM-1)

typedef __attribute__((ext_vector_type(2))) float v2f;
typedef __attribute__((ext_vector_type(8))) float v8f;

// ---------------------------------------------------------------------------
// Compile-time Cl(3,0) algebra tables.
// Blade order: (), e0, e1, e2, e01, e02, e12, e012  -> masks {0,1,2,4,3,5,6,7}
// (idx<->mask is an involution: 3<->4, rest fixed).
// path[] follows np.argwhere(paths) row-major order over (gi,gj,gk), 20 paths.
// ---------------------------------------------------------------------------
struct Alg {
    float sign[NB][NB];   // sign of blade_i * blade_k   (metric = +1,+1,+1)
    int   jidx[NB][NB];   // blade index of the product  (xor of masks)
    int   path[4][4][4];  // grade-triple -> path index, -1 if not a path
    int   grade[NB];
};

constexpr int popc3(int x) { return (x & 1) + ((x >> 1) & 1) + ((x >> 2) & 1); }

constexpr Alg make_alg() {
    Alg A{};
    int mask[NB] = {0, 1, 2, 4, 3, 5, 6, 7};
    for (int i = 0; i < NB; i++) A.grade[i] = popc3(mask[i]);
    for (int i = 0; i < NB; i++)
        for (int k = 0; k < NB; k++) {
            int ma = mask[i], mb = mask[k];
            int s = 0, t = ma >> 1;
            while (t) { s += popc3(t & mb); t >>= 1; }
            A.sign[i][k] = (s & 1) ? -1.0f : 1.0f;
            A.jidx[i][k] = mask[ma ^ mb];   // involution: mask->idx
        }
    int p = 0;
    for (int gi = 0; gi < 4; gi++)
        for (int gj = 0; gj < 4; gj++)
            for (int gk = 0; gk < 4; gk++) {
                int lo  = gi > gk ? gi - gk : gk - gi;
                int hi1 = gi + gk, hi2 = 6 - (gi + gk);
                int hi  = hi1 < hi2 ? hi1 : hi2;
                bool ex = (((gi + gj + gk) & 1) == 0) && gj >= lo && gj <= hi;
                A.path[gi][gj][gk] = ex ? p++ : -1;
            }
    return A;
}

__constant__ Alg g_alg = make_alg();

__device__ __forceinline__ float sigmoidf(float z) { return 1.0f / (1.0f + expf(-z)); }

// ---------------------------------------------------------------------------
// Kernel 0: build initial multivector field x (343,1,8) and q_pos.
// X layout for layer 0: [blade][MP][Kpad=4], k=0 holds the value, k=1..3 zero.
// ---------------------------------------------------------------------------
__global__ void init_x_kernel(const float* __restrict__ sig,
                              float* __restrict__ X, float* __restrict__ qpos) {
    int n = blockIdx.x * blockDim.x + threadIdx.x;
    if (n >= MP) return;
    float val[NB] = {0, 0, 0, 0, 0, 0, 0, 0};
    float q = 0.f;
    if (n < NPOS) {
        int a = n / 49, b = (n / 7) % 7, c = n % 7;
        float ra = (a - 3) / 3.0f, rb = (b - 3) / 3.0f, rc = (c - 3) / 3.0f;
        q = ra * ra + rb * rb + rc * rc;
        float s = sig[0];
        val[0] = expf(-q / (s * s));
        val[1] = ra; val[2] = rb; val[3] = rc;
    }
    qpos[n] = q;
    for (int i = 0; i < NB; i++) {
        X[((size_t)i * MP + n) * 4 + 0] = val[i];
        X[((size_t)i * MP + n) * 4 + 1] = 0.f;
        X[((size_t)i * MP + n) * 4 + 2] = 0.f;
        X[((size_t)i * MP + n) * 4 + 3] = 0.f;
    }
}

// ---------------------------------------------------------------------------
// Pack weight (N, Kreal, 4) row-major into WMMA-fragment order, zero-padded:
//   Wp[g][Kpad/4][half][N][2],  element (g,k4,half,n,j) = W[n][k4*4+half*2+j][g]
// Each GEMM lane then loads its B (k,k+1) pair as ONE contiguous b64, and the
// 16 lanes of a fragment half cover a contiguous 128B line.
// ---------------------------------------------------------------------------
__global__ void pack_wt_kernel(const float* __restrict__ W, float* __restrict__ Wt,
                               int Kreal, int Kpad, int N) {
    int t = blockIdx.x * blockDim.x + threadIdx.x;
    int total = 4 * Kpad * N;
    if (t >= total) return;
    int j    = t & 1;
    int n    = (t >> 1) % N;
    int half = (t / (2 * N)) & 1;
    int k4   = (t / (4 * N)) % (Kpad >> 2);
    int g    = t / (Kpad * N);
    int k    = k4 * 4 + half * 2 + j;
    Wt[t] = (k < Kreal) ? W[((size_t)n * Kreal + k) * 4 + g] : 0.f;
}

// ---------------------------------------------------------------------------
// WMMA GEMM per blade: Y[blade][m][n] = sum_k A[blade][m][k] * W[n][k][grade]
// (+ bias[n] on blade 0).
// Block = 128 threads = 4 waves in a 2(M) x 2(N) arrangement covering 32x64.
// Each wave owns a 16x32 tile: TWO 16x16 accumulators sharing one A fragment,
// so each K-step is 3 unconditional b64 loads feeding 2 V_WMMA_F32_16X16X4_F32.
// A frag: lanes 0-15 M rows / k0,k0+1; lanes 16-31 / k0+2,k0+3.
// D frag: VGPR r -> M = r + 8*(lane>>4), N = lane&15.
// ---------------------------------------------------------------------------
__global__ void wmma_gemm_blades(const float* __restrict__ A8,  // [NB][MP][K]
                                 const float* __restrict__ Wt,  // packed, see above
                                 const float* __restrict__ bias,// [N] or null
                                 float* __restrict__ Y8,        // [NB][MP][N]
                                 int K, int N) {
    int blade = blockIdx.z;
    int g = g_alg.grade[blade];
    const float* A = A8 + (size_t)blade * MP * K;
    const float* B = Wt + (size_t)g * K * N;
    float*       Y = Y8 + (size_t)blade * MP * N;

    int wave = threadIdx.x >> 5;
    int lane = threadIdx.x & 31;
    int half = lane >> 4;        // 0 or 1 (selects K sub-pair)
    int lid  = lane & 15;
    int m0 = blockIdx.y * 32 + (wave >> 1) * 16;
    int n0 = blockIdx.x * 64 + (wave & 1) * 32;

    const float* Arow = A + (size_t)(m0 + lid) * K + half * 2;          // 8B aligned
    // packed B: element (k0,half,ncol,j) at offset k0*N + half*2N + ncol*2 + j
    const float* Bp   = B + (size_t)half * (2 * N) + (size_t)(n0 + lid) * 2;

    v8f acc0 = {0.f, 0.f, 0.f, 0.f, 0.f, 0.f, 0.f, 0.f};
    v8f acc1 = {0.f, 0.f, 0.f, 0.f, 0.f, 0.f, 0.f, 0.f};
#pragma unroll 4
    for (int k0 = 0; k0 < K; k0 += 4) {
        v2f a  = *(const v2f*)(Arow + k0);
        v2f b0 = *(const v2f*)(Bp + (size_t)k0 * N);        // cols n0..n0+15
        v2f b1 = *(const v2f*)(Bp + (size_t)k0 * N + 32);   // cols n0+16..n0+31
        acc0 = __builtin_amdgcn_wmma_f32_16x16x4_f32(false, a, false, b0,
                                                     (short)0, acc0, false, false);
        acc1 = __builtin_amdgcn_wmma_f32_16x16x4_f32(false, a, false, b1,
                                                     (short)0, acc1, false, false);
    }
    bool bb = (bias != nullptr) && (blade == 0);
    float bv0 = bb ? bias[n0 + lid]      : 0.f;
    float bv1 = bb ? bias[n0 + 16 + lid] : 0.f;
#pragma unroll
    for (int r = 0; r < 8; r++) {
        int mr = m0 + r + half * 8;
        Y[(size_t)mr * N + n0 + lid]      = acc0[r] + bv0;
        Y[(size_t)mr * N + n0 + 16 + lid] = acc1[r] + bv1;
    }
}

// ---------------------------------------------------------------------------
// mv_silu: gate[g] = sigmoid(a*inv + b); inv = [x0, |x|_g1, |x|_g2, |x|_g3]
// ---------------------------------------------------------------------------
__global__ void mv_silu_kernel(float* __restrict__ Y,
                               const float* __restrict__ sa,
                               const float* __restrict__ sb, int C) {
    int t = blockIdx.x * blockDim.x + threadIdx.x;
    if (t >= NPOS * C) return;
    int n = t / C, c = t - n * C;
    size_t base = (size_t)n * C + c, str = (size_t)MP * C;
    float v[NB];
    for (int i = 0; i < NB; i++) v[i] = Y[i * str + base];
    float q1 = v[1]*v[1] + v[2]*v[2] + v[3]*v[3];
    float q2 = v[4]*v[4] + v[5]*v[5] + v[6]*v[6];
    float q3 = v[7]*v[7];
    float inv[4] = { v[0], sqrtf(q1 + SQ_EPSF), sqrtf(q2 + SQ_EPSF), sqrtf(q3 + SQ_EPSF) };
    float gate[4];
    for (int g = 0; g < 4; g++)
        gate[g] = sigmoidf(sa[c * 4 + g] * inv[g] + sb[c * 4 + g]);
    for (int i = 0; i < NB; i++) Y[i * str + base] = gate[g_alg.grade[i]] * v[i];
}

// ---------------------------------------------------------------------------
// mv_normalize: n_g = sigmoid(a)*(sqrt(q_g+eps)-1)+1; x /= (n_g + EPS)
// ---------------------------------------------------------------------------
__global__ void mv_norm_kernel(float* __restrict__ XR,
                               const float* __restrict__ na, int C) {
    int t = blockIdx.x * blockDim.x + threadIdx.x;
    if (t >= NPOS * C) return;
    int n = t / C, c = t - n * C;
    size_t base = (size_t)n * C + c, str = (size_t)MP * C;
    float v[NB];
    for (int i = 0; i < NB; i++) v[i] = XR[i * str + base];
    float q[4];
    q[0] = v[0]*v[0];
    q[1] = v[1]*v[1] + v[2]*v[2] + v[3]*v[3];
    q[2] = v[4]*v[4] + v[5]*v[5] + v[6]*v[6];
    q[3] = v[7]*v[7];
    float nrm[4];
    for (int g = 0; g < 4; g++) {
        float nn = sqrtf(q[g] + SQ_EPSF);
        nrm[g] = sigmoidf(na[c * 4 + g]) * (nn - 1.0f) + 1.0f;
    }
    for (int i = 0; i < NB; i++) XR[i * str + base] = v[i] / (nrm[g_alg.grade[i]] + EPSF);
}

// ---------------------------------------------------------------------------
// sgp combine: gp[j] = sum_{i,k} sign(i,k)*gpw[c,path(gi,gj,gk)]*x[i]*xr[k];
// out = (wl_out + gp) / sqrt(2).  Pad rows (n >= NPOS) are zeroed so the next
// layer's GEMM can load them unconditionally.
// ---------------------------------------------------------------------------
__global__ void sgp_combine_kernel(const float* __restrict__ Xs,   // post-silu
                                   const float* __restrict__ XR,   // normalized
                                   const float* __restrict__ Lc,   // wl output
                                   const float* __restrict__ gpw,  // [C][20]
                                   float* __restrict__ Xout, int C) {
    int t = blockIdx.x * blockDim.x + threadIdx.x;
    if (t >= MP * C) return;
    int n = t / C, c = t - n * C;
    size_t base = (size_t)n * C + c, str = (size_t)MP * C;
    if (n >= NPOS) {   // zero pad rows
        for (int j = 0; j < NB; j++) Xout[j * str + base] = 0.f;
        return;
    }
    float v[NB], xr[NB], l[NB], pw[PSUM];
    for (int i = 0; i < NB; i++) {
        v[i]  = Xs[i * str + base];
        xr[i] = XR[i * str + base];
        l[i]  = Lc[i * str + base];
    }
    for (int p = 0; p < PSUM; p++) pw[p] = gpw[(size_t)c * PSUM + p];
    float gp[NB] = {0, 0, 0, 0, 0, 0, 0, 0};
    for (int i = 0; i < NB; i++)
        for (int k = 0; k < NB; k++) {
            int j  = g_alg.jidx[i][k];
            int p  = g_alg.path[g_alg.grade[i]][g_alg.grade[j]][g_alg.grade[k]];
            gp[j] += g_alg.sign[i][k] * pw[p] * v[i] * xr[k];
        }
    const float is2 = 0.70710678118654752f;
    for (int j = 0; j < NB; j++) Xout[j * str + base] = (l[j] + gp[j]) * is2;
}

// ---------------------------------------------------------------------------
// channel layer norm: x = ln_a[c] * x / (mean_c sqrt(sum_i x^2 + eps) + EPS)
// one block per position n.  Only n < NPOS rows touched (pad rows stay zero).
// ---------------------------------------------------------------------------
__global__ void ln_kernel(float* __restrict__ X, const float* __restrict__ lna, int C) {
    int n = blockIdx.x;
    __shared__ float red[256];
    __shared__ float s_mean;
    size_t str = (size_t)MP * C;
    float acc = 0.f;
    for (int c = threadIdx.x; c < C; c += blockDim.x) {
        float s = 0.f;
        for (int i = 0; i < NB; i++) {
            float v = X[i * str + (size_t)n * C + c];
            s += v * v;
        }
        acc += sqrtf(s + SQ_EPSF);
    }
    red[threadIdx.x] = acc;
    __syncthreads();
    for (int w = 128; w > 0; w >>= 1) {
        if (threadIdx.x < w) red[threadIdx.x] += red[threadIdx.x + w];
        __syncthreads();
    }
    if (threadIdx.x == 0) s_mean = red[0] / (float)C;
    __syncthreads();
    float denom = s_mean + EPSF;
    for (int c = threadIdx.x; c < C; c += blockDim.x) {
        float sc = lna[c] / denom;
        for (int i = 0; i < NB; i++) X[i * str + (size_t)n * C + c] *= sc;
    }
}

// ---------------------------------------------------------------------------
// final: k = x * exp(-q*shell_sigma) * FACTOR ;  K[o,l,i,m,n] = sum_k k_k *
// sign(k,m) * cw[c, path(gk,gl,gm)] with l = xor(k,m)
// out[(o*8+l)*256 + i*8+m][n]
// ---------------------------------------------------------------------------
__global__ void final_kernel(const float* __restrict__ X,      // [NB][MP][1024]
                             const float* __restrict__ qpos,
                             const float* __restrict__ ssig,   // [1024][8]
                             const float* __restrict__ cw,     // [1024][20]
                             float* __restrict__ out) {
    int n = blockIdx.x * blockDim.x + threadIdx.x;
    int c = blockIdx.y;                 // c = o*32 + i
    if (n >= NPOS) return;
    const int C = 1024;
    int o = c >> 5, ii = c & 31;
    size_t str = (size_t)MP * C;
    float q = qpos[n];
    float kv[NB];
    for (int b = 0; b < NB; b++)
        kv[b] = X[b * str + (size_t)n * C + c] * expf(-q * ssig[c * NB + b]) * FACTORF;
    float pw[PSUM];
    for (int p = 0; p < PSUM; p++) pw[p] = cw[(size_t)c * PSUM + p];
    float ov[NB][NB];
    for (int l = 0; l < NB; l++)
        for (int m = 0; m < NB; m++) ov[l][m] = 0.f;
    for (int kb = 0; kb < NB; kb++)
        for (int m = 0; m < NB; m++) {
            int l = g_alg.jidx[kb][m];
            int p = g_alg.path[g_alg.grade[kb]][g_alg.grade[l]][g_alg.grade[m]];
            ov[l][m] += kv[kb] * g_alg.sign[kb][m] * pw[p];
        }
    for (int l = 0; l < NB; l++)
        for (int m = 0; m < NB; m++)
            out[((size_t)(o * NB + l) * 256 + (size_t)(ii * NB + m)) * NPOS + n] = ov[l][m];
}

// ---------------------------------------------------------------------------
// host side
// ---------------------------------------------------------------------------
static void run_gemm(const float* A, const float* W, float* Wt, const float* bias,
                     float* Y, int Kreal, int Kpad, int N, hipStream_t s) {
    int tot = 4 * Kpad * N;
    pack_wt_kernel<<<(tot + 255) / 256, 256, 0, s>>>(W, Wt, Kreal, Kpad, N);
    dim3 grid(N / 64, MP / 32, NB);
    wmma_gemm_blades<<<grid, 128, 0, s>>>(A, Wt, bias, Y, Kpad, N);
}

extern "C" void kernel_launch(void* const* d_in, const int* in_sizes, int n_in,
                              void* d_out, int out_size, void* d_ws, size_t ws_size,
                              hipStream_t stream) {
    (void)out_size; (void)ws_size;
    const float *rel_sigma = nullptr, *shell = nullptr, *cw = nullptr;
    struct LP { const float *w, *b, *sa, *sb, *wl, *bl, *wr, *na, *gpw, *ln; } L[4];
    auto F = [&](int i) { return (const float*)d_in[i]; };

    if (n_in >= 43 && in_sizes[0] == 1) {
        // dict insertion order
        int i = 0;
        rel_sigma = F(i++); shell = F(i++); cw = F(i++);
        for (int l = 0; l < 4; l++) {
            L[l].w  = F(i++); L[l].b  = F(i++); L[l].sa = F(i++); L[l].sb = F(i++);
            L[l].wl = F(i++); L[l].bl = F(i++); L[l].wr = F(i++); L[l].na = F(i++);
            L[l].gpw = F(i++); L[l].ln = F(i++);
        }
    } else {
        // jax sorted-pytree order: cayley_weight, layers(sorted keys), rel_pos_sigma, shell_sigma
        int i = 0;
        cw = F(i++);
        for (int l = 0; l < 4; l++) {
            L[l].b  = F(i++); L[l].ln = F(i++); L[l].bl = F(i++); L[l].gpw = F(i++);
            L[l].na = F(i++); L[l].wl = F(i++); L[l].wr = F(i++); L[l].sa  = F(i++);
            L[l].sb = F(i++); L[l].w  = F(i++);
        }
        rel_sigma = F(i++); shell = F(i++);
    }

    float* ws = (float*)d_ws;
    const size_t BUF = (size_t)NB * MP * 1024;
    float* qpos = ws;
    float* X    = ws + 512;
    float* Y    = X + BUF;
    float* XR   = Y + BUF;
    float* Lb   = XR + BUF;
    float* Wt   = Lb + BUF;            // 4 * 1024 * 1024 floats (16 MB), reused

    init_x_kernel<<<dim3((MP + 127) / 128), 128, 0, stream>>>(rel_sigma, X, qpos);

    const int dims[5] = {1, 64, 64, 64, 1024};
    for (int l = 0; l < 4; l++) {
        int ci = dims[l], co = dims[l + 1];
        int ci_pad = (l == 0) ? 4 : ci;          // layer 0: K padded 1 -> 4
        int tot = NPOS * co;
        run_gemm(X, L[l].w, Wt, L[l].b, Y, ci, ci_pad, co, stream);        // mv_linear
        mv_silu_kernel<<<(tot + 255) / 256, 256, 0, stream>>>(Y, L[l].sa, L[l].sb, co);
        run_gemm(Y, L[l].wr, Wt, nullptr, XR, co, co, co, stream);         // sgp wr
        mv_norm_kernel<<<(tot + 255) / 256, 256, 0, stream>>>(XR, L[l].na, co);
        run_gemm(Y, L[l].wl, Wt, L[l].bl, Lb, co, co, co, stream);         // sgp wl
        sgp_combine_kernel<<<(MP * co + 255) / 256, 256, 0, stream>>>(Y, XR, Lb, L[l].gpw, X, co);
        ln_kernel<<<NPOS, 256, 0, stream>>>(X, L[l].ln, co);
    }

    dim3 fg((NPOS + 127) / 128, 1024);
    final_kernel<<<fg, 128, 0, stream>>>(X, qpos, shell, cw, (float*)d_out);
}